// ScorePredictor_24721831756410
// MI455X (gfx1250) — compile-verified
//
#include <hip/hip_runtime.h>

typedef __attribute__((ext_vector_type(2))) float v2f;
typedef __attribute__((ext_vector_type(8))) float v8f;

#define DIM 128

// One wave32 handles 16 edges: A = gathered hu rows (M=edge), B = gathered
// hv rows scaled by r (N=edge). 32 chained V_WMMA_F32_16X16X4_F32 build the
// 16x16 Gram tile; scores are its diagonal.
__global__ __launch_bounds__(256) void ScorePredictor_edge_dot_wmma(
    const float* __restrict__ h,
    const float* __restrict__ r,
    const int*   __restrict__ src_idx,
    const int*   __restrict__ dst_idx,
    float* __restrict__ out,
    int n_edges)
{
    const unsigned lane = threadIdx.x & 31u;
    const unsigned wave = threadIdx.x >> 5;
    const int batch = (int)blockIdx.x * 8 + (int)wave;   // 16 edges per batch
    const int base  = batch * 16;
    if (base >= n_edges) return;

    // Edge owned by this lane (both wave halves cover the same 16 edges).
    int e = base + (int)(lane & 15u);
    if (e > n_edges - 1) e = n_edges - 1;                // clamp tail (stores masked below)

    const long s = (long)src_idx[e];
    const long d = (long)dst_idx[e];
    const float* __restrict__ pa = h + s * DIM;          // A row (hu)
    const float* __restrict__ pb = h + d * DIM;          // B row (hv)

    // fp32 WMMA A/B layout: lanes 0-15 carry K=k,k+1 ; lanes 16-31 carry K=k+2,k+3.
    const int koff = (lane >= 16u) ? 2 : 0;

    v8f c = {};
    #pragma unroll
    for (int k = 0; k < DIM; k += 4) {
        v2f a  = *(const v2f*)(pa + k + koff);           // global_load_b64 -> A VGPR pair
        v2f b  = *(const v2f*)(pb + k + koff);           // global_load_b64 -> B VGPR pair
        v2f rr = *(const v2f*)(r  + k + koff);           // broadcast, L0-resident
        b = b * rr;                                      // fold r into B
        // D = A(16x4) x B(4x16) + C : v_wmma_f32_16x16x4_f32
        c = __builtin_amdgcn_wmma_f32_16x16x4_f32(
                /*neg_a=*/false, a, /*neg_b=*/false, b,
                /*c_mod=*/(short)0, c, /*reuse_a=*/false, /*reuse_b=*/false);
    }

    // Diagonal extraction from C/D layout:
    //   edge n in [0,8):  lane n,     VGPR n
    //   edge n in [8,16): lane n+16,  VGPR n-8
    float val = 0.0f;
    #pragma unroll
    for (int j = 0; j < 8; ++j) {
        if (lane == (unsigned)j)        val = c[j];
        if (lane == (unsigned)(j + 24)) val = c[j];
    }
    int eib = (lane < 8u) ? (int)lane : ((lane >= 24u) ? ((int)lane - 16) : -1);
    if (eib >= 0) {
        int eo = base + eib;
        if (eo < n_edges) out[eo] = val;
    }
}

extern "C" void kernel_launch(void* const* d_in, const int* in_sizes, int n_in,
                              void* d_out, int out_size, void* d_ws, size_t ws_size,
                              hipStream_t stream) {
    const float* h   = (const float*)d_in[0];
    const float* r   = (const float*)d_in[1];
    const int*   src = (const int*)d_in[2];
    const int*   dst = (const int*)d_in[3];
    float* out = (float*)d_out;

    const int n_edges   = in_sizes[2];
    const int n_batches = (n_edges + 15) / 16;   // 16 edges per wave
    const int blocks    = (n_batches + 7) / 8;   // 8 waves (256 threads) per block

    ScorePredictor_edge_dot_wmma<<<blocks, 256, 0, stream>>>(h, r, src, dst, out, n_edges);
}